// CustomNetwork_19189913879215
// MI455X (gfx1250) — compile-verified
//
#include <hip/hip_runtime.h>
#include <math.h>

typedef __attribute__((ext_vector_type(2))) float v2f;
typedef __attribute__((ext_vector_type(4))) float v4f;
typedef __attribute__((ext_vector_type(8))) float v8f;

#define BB   1024
#define NN   256
#define FIN  6
#define HD   32
#define EMBD 32
#define VECD 26
#define DPI  512

// ---------------- graph preprocessing: dense normalized adjacency ----------------

__global__ void zero_f32(float* p, int n) {
    int i = blockIdx.x * 256 + threadIdx.x;
    if (i < n) p[i] = 0.0f;
}

__global__ void degree_kernel(const int* __restrict__ src, const int* __restrict__ dst,
                              float* __restrict__ deg_out, float* __restrict__ deg_in, int E) {
    int e = blockIdx.x * 256 + threadIdx.x;
    if (e < E) {
        atomicAdd(&deg_out[src[e]], 1.0f);
        atomicAdd(&deg_in[dst[e]], 1.0f);
    }
}

__global__ void build_A_kernel(const int* __restrict__ src, const int* __restrict__ dst,
                               const float* __restrict__ deg_out, const float* __restrict__ deg_in,
                               float* __restrict__ A, int E) {
    int e = blockIdx.x * 256 + threadIdx.x;
    if (e < E) {
        int s = src[e], d = dst[e];
        float go = deg_out[s], gi = deg_in[d];
        float io = go > 0.0f ? 1.0f / sqrtf(go) : 0.0f;
        float ii = gi > 0.0f ? 1.0f / sqrtf(gi) : 0.0f;
        atomicAdd(&A[d * NN + s], io * ii);
    }
}

// ---------------- WMMA helpers (f32 16x16x4) ----------------
// A operand (16x4 f32): lane L holds row M=L%16; VGPR0=K(kk), VGPR1=K(kk+1), kk=k+2*(L/16)
// B operand (4x16 f32): lane L holds col N=L%16; VGPR0=row kk, VGPR1=row kk+1
// C/D (16x16 f32): vgpr v, lane L -> row = v + 8*(L/16), col = L%16

// H_out[m, n] = relu( sum_k A[m,k] * P[k,n] + bias[n] ), per-wave 32 rows
__device__ __forceinline__ void spmm_relu(const float* __restrict__ sA,
                                          const float* __restrict__ sP,
                                          float* __restrict__ sHout,
                                          const float* __restrict__ bias,
                                          int wave, int lane) {
    const int l15 = lane & 15;
    const int lhi = lane >> 4;
    const int m0 = wave * 32;

    v8f c00 = {}, c01 = {}, c10 = {}, c11 = {};
    const float* a0base = sA + (m0 + l15) * NN;
    const float* a1base = sA + (m0 + 16 + l15) * NN;

#pragma unroll 4
    for (int k = 0; k < NN; k += 4) {
        const int kk = k + 2 * lhi;
        v2f a0 = { a0base[kk], a0base[kk + 1] };
        v2f a1 = { a1base[kk], a1base[kk + 1] };
        v2f p0 = { sP[kk * HD + l15],      sP[(kk + 1) * HD + l15] };
        v2f p1 = { sP[kk * HD + 16 + l15], sP[(kk + 1) * HD + 16 + l15] };
        c00 = __builtin_amdgcn_wmma_f32_16x16x4_f32(false, a0, false, p0, (short)0, c00, false, false);
        c01 = __builtin_amdgcn_wmma_f32_16x16x4_f32(false, a0, false, p1, (short)0, c01, false, false);
        c10 = __builtin_amdgcn_wmma_f32_16x16x4_f32(false, a1, false, p0, (short)0, c10, false, false);
        c11 = __builtin_amdgcn_wmma_f32_16x16x4_f32(false, a1, false, p1, (short)0, c11, false, false);
    }

    const float bias0 = bias[l15];
    const float bias1 = bias[16 + l15];
#pragma unroll
    for (int v = 0; v < 8; v++) {
        int r0 = m0 + v + 8 * lhi;
        int r1 = r0 + 16;
        sHout[r0 * HD + l15]      = fmaxf(c00[v] + bias0, 0.0f);
        sHout[r0 * HD + 16 + l15] = fmaxf(c01[v] + bias1, 0.0f);
        sHout[r1 * HD + l15]      = fmaxf(c10[v] + bias0, 0.0f);
        sHout[r1 * HD + 16 + l15] = fmaxf(c11[v] + bias1, 0.0f);
    }
}

// P[m, n] = sum_k H[m,k] * W[k,n]   (K = 32, W in global memory, row-major [32,32])
__device__ __forceinline__ void dense_hw(const float* __restrict__ sH,
                                         const float* __restrict__ W,
                                         float* __restrict__ sPout,
                                         int wave, int lane) {
    const int l15 = lane & 15;
    const int lhi = lane >> 4;
    const int m0 = wave * 32;

    v8f c00 = {}, c01 = {}, c10 = {}, c11 = {};
    const float* h0base = sH + (m0 + l15) * HD;
    const float* h1base = sH + (m0 + 16 + l15) * HD;

#pragma unroll
    for (int k = 0; k < HD; k += 4) {
        const int kk = k + 2 * lhi;
        v2f a0 = { h0base[kk], h0base[kk + 1] };
        v2f a1 = { h1base[kk], h1base[kk + 1] };
        v2f w0 = { W[kk * HD + l15],      W[(kk + 1) * HD + l15] };
        v2f w1 = { W[kk * HD + 16 + l15], W[(kk + 1) * HD + 16 + l15] };
        c00 = __builtin_amdgcn_wmma_f32_16x16x4_f32(false, a0, false, w0, (short)0, c00, false, false);
        c01 = __builtin_amdgcn_wmma_f32_16x16x4_f32(false, a0, false, w1, (short)0, c01, false, false);
        c10 = __builtin_amdgcn_wmma_f32_16x16x4_f32(false, a1, false, w0, (short)0, c10, false, false);
        c11 = __builtin_amdgcn_wmma_f32_16x16x4_f32(false, a1, false, w1, (short)0, c11, false, false);
    }

#pragma unroll
    for (int v = 0; v < 8; v++) {
        int r0 = m0 + v + 8 * lhi;
        int r1 = r0 + 16;
        sPout[r0 * HD + l15]      = c00[v];
        sPout[r0 * HD + 16 + l15] = c01[v];
        sPout[r1 * HD + l15]      = c10[v];
        sPout[r1 * HD + 16 + l15] = c11[v];
    }
}

// ---------------- fused GNN forward ----------------

__global__ __launch_bounds__(256, 1)
void gnn_forward(const float* __restrict__ gfeat, const float* __restrict__ vec,
                 const float* __restrict__ W1, const float* __restrict__ b1,
                 const float* __restrict__ W2, const float* __restrict__ b2,
                 const float* __restrict__ W3, const float* __restrict__ b3,
                 const float* __restrict__ Wemb, const float* __restrict__ bemb,
                 const float* __restrict__ Wpi, const float* __restrict__ bpi,
                 const float* __restrict__ Wvf, const float* __restrict__ bvf,
                 const float* __restrict__ A, float* __restrict__ out) {
    __shared__ float sA[NN * NN];   // 256 KB dense normalized adjacency
    __shared__ float sH[NN * HD];   // 32 KB hidden
    __shared__ float sP[NN * HD];   // 32 KB projected / scratch

    const int tid  = threadIdx.x;
    const int wave = tid >> 5;
    const int lane = tid & 31;

    // Load A into LDS once per workgroup (vectorized)
    {
        const v4f* s4 = (const v4f*)A;
        v4f* d4 = (v4f*)sA;
        for (int i = tid; i < (NN * NN) / 4; i += 256) d4[i] = s4[i];
    }
    __syncthreads();

    for (int b = blockIdx.x; b < BB; b += gridDim.x) {
        // ---- Layer 1 projection: P1 = X @ W1  (X: [256,6], W1: [6,32]) ----
        if (tid < FIN * HD) sH[tid] = W1[tid];   // stage W1 (sH free right now)
        __syncthreads();
        {
            const float* xr = gfeat + ((size_t)b * NN + tid) * FIN;
            float x[FIN];
#pragma unroll
            for (int k = 0; k < FIN; k++) x[k] = xr[k];
#pragma unroll
            for (int c = 0; c < HD; c++) {
                float s = 0.0f;
#pragma unroll
                for (int k = 0; k < FIN; k++) s += x[k] * sH[k * HD + c];
                sP[tid * HD + c] = s;
            }
        }
        __syncthreads();

        spmm_relu(sA, sP, sH, b1, wave, lane);   // H1 = relu(A@P1 + b1)
        __syncthreads();
        dense_hw(sH, W2, sP, wave, lane);        // P2 = H1 @ W2
        __syncthreads();
        spmm_relu(sA, sP, sH, b2, wave, lane);   // H2 = relu(A@P2 + b2)
        __syncthreads();
        dense_hw(sH, W3, sP, wave, lane);        // P3 = H2 @ W3
        __syncthreads();
        spmm_relu(sA, sP, sH, b3, wave, lane);   // H3 = relu(A@P3 + b3)
        __syncthreads();

        // ---- mean-pool (all 8 waves), embedding, concat (sP as scratch) ----
        float* sHg   = sP;         // 32 floats
        float* sComb = sP + 64;    // 58 floats
        float* sPart = sP + 128;   // 8 * 32 partial sums
        {
            const int c = lane;    // column 0..31
            float s = 0.0f;
            const int r0 = wave * 32;
#pragma unroll 8
            for (int r = 0; r < 32; r++) s += sH[(r0 + r) * HD + c];
            sPart[wave * HD + c] = s;
        }
        __syncthreads();
        if (tid < HD) {
            float s = 0.0f;
#pragma unroll
            for (int w = 0; w < 8; w++) s += sPart[w * HD + tid];
            sHg[tid] = s * (1.0f / NN);
        }
        __syncthreads();
        if (tid < EMBD) {
            float e = bemb[tid];
#pragma unroll
            for (int k = 0; k < HD; k++) e += sHg[k] * Wemb[k * EMBD + tid];
            sComb[VECD + tid] = e;
        }
        if (tid >= 32 && tid < 32 + VECD) {
            sComb[tid - 32] = vec[(size_t)b * VECD + (tid - 32)];
        }
        __syncthreads();

        // ---- heads: pi/vf = relu(comb @ W + b), each [58] x [58,512] ----
#pragma unroll
        for (int jj = 0; jj < 2; jj++) {
            const int j = tid + jj * 256;
            float spi = bpi[j];
            float svf = bvf[j];
            for (int k = 0; k < VECD + EMBD; k++) {
                const float cv = sComb[k];
                spi += cv * Wpi[k * DPI + j];
                svf += cv * Wvf[k * DPI + j];
            }
            out[(size_t)b * DPI + j]                    = fmaxf(spi, 0.0f);
            out[(size_t)BB * DPI + (size_t)b * DPI + j] = fmaxf(svf, 0.0f);
        }
        __syncthreads();   // protect sH/sP before next batch iteration
    }
}

// ---------------- launcher ----------------

extern "C" void kernel_launch(void* const* d_in, const int* in_sizes, int n_in,
                              void* d_out, int out_size, void* d_ws, size_t ws_size,
                              hipStream_t stream) {
    const float* gfeat = (const float*)d_in[0];
    const float* vec   = (const float*)d_in[1];
    const int*   src   = (const int*)d_in[2];
    const int*   dst   = (const int*)d_in[3];
    const float* W1    = (const float*)d_in[4];
    const float* b1    = (const float*)d_in[5];
    const float* W2    = (const float*)d_in[6];
    const float* b2    = (const float*)d_in[7];
    const float* W3    = (const float*)d_in[8];
    const float* b3    = (const float*)d_in[9];
    const float* Wemb  = (const float*)d_in[10];
    const float* bemb  = (const float*)d_in[11];
    const float* Wpi   = (const float*)d_in[12];
    const float* bpi   = (const float*)d_in[13];
    const float* Wvf   = (const float*)d_in[14];
    const float* bvf   = (const float*)d_in[15];

    float* A       = (float*)d_ws;          // [256*256]
    float* deg_out = A + NN * NN;           // [256]
    float* deg_in  = deg_out + NN;          // [256]
    const int E = in_sizes[2];

    const int ztotal = NN * NN + 2 * NN;
    zero_f32<<<(ztotal + 255) / 256, 256, 0, stream>>>(A, ztotal);
    degree_kernel<<<(E + 255) / 256, 256, 0, stream>>>(src, dst, deg_out, deg_in, E);
    build_A_kernel<<<(E + 255) / 256, 256, 0, stream>>>(src, dst, deg_out, deg_in, A, E);

    gnn_forward<<<256, 256, 0, stream>>>(gfeat, vec, W1, b1, W2, b2, W3, b3,
                                         Wemb, bemb, Wpi, bpi, Wvf, bvf,
                                         A, (float*)d_out);
}